// V6LayerTrueFast_39900246180532
// MI455X (gfx1250) — compile-verified
//
#include <hip/hip_runtime.h>
#include <math.h>

#define Bx   4
#define Tt   2048
#define Cc   768
#define Mm   (Bx*Tt)
#define DKk  96
#define Nanc 8
#define CHk  128
#define NCH  (Tt/CHk)

typedef __attribute__((ext_vector_type(2))) float v2f;
typedef __attribute__((ext_vector_type(8))) float v8f;

// ---------------------------------------------------------------------------
// Fused WMMA (fp32, 16x16x4) GEMM:  Y = epilogue( X[M,K] @ W[N,K]^T )
// act: 0=none, 1=tanh, 2=sigmoid.  val = act(acc + bias[col] + addend[r,c]) * mul[r,c]
// ---------------------------------------------------------------------------
__global__ __launch_bounds__(256)
void gemm_wmma_f32(const float* __restrict__ X, const float* __restrict__ W,
                   const float* __restrict__ bias, const float* __restrict__ addend,
                   const float* __restrict__ mul, float* __restrict__ Y,
                   int Mrows, int Nout, int Kdim, int act)
{
    __shared__ float As[128 * 33];
    __shared__ float Bs[64 * 33];

    const int tid   = threadIdx.x;
    const int row0  = blockIdx.y * 128;
    const int col0  = blockIdx.x * 64;
    const int lane  = tid & 31;
    const int wid   = tid >> 5;
    const int wr    = (wid & 3) * 32;   // wave row offset in tile
    const int wc    = (wid >> 2) * 32;  // wave col offset in tile
    const int lidx  = lane & 15;
    const int lhalf = lane >> 4;        // 0: K=k,k+1   1: K=k+2,k+3

    v8f acc00 = {}; v8f acc01 = {}; v8f acc10 = {}; v8f acc11 = {};

    for (int k0 = 0; k0 < Kdim; k0 += 32) {
        __syncthreads();
        // stage X tile 128x32 (coalesced, padded stride 33)
        #pragma unroll
        for (int i = 0; i < 16; ++i) {
            int idx = tid + i * 256;
            int r = idx >> 5, kk = idx & 31;
            int gr = row0 + r;
            As[r * 33 + kk] = (gr < Mrows) ? X[(size_t)gr * Kdim + k0 + kk] : 0.f;
        }
        // stage W tile 64x32
        #pragma unroll
        for (int i = 0; i < 8; ++i) {
            int idx = tid + i * 256;
            int n = idx >> 5, kk = idx & 31;
            int gn = col0 + n;
            Bs[n * 33 + kk] = (gn < Nout) ? W[(size_t)gn * Kdim + k0 + kk] : 0.f;
        }
        if (k0 + 32 < Kdim)  // hint next A tile into cache (global_prefetch)
            __builtin_prefetch(&X[(size_t)(row0 + (tid >> 1)) * Kdim + k0 + 32 + (tid & 1) * 16], 0, 1);
        __syncthreads();

        #pragma unroll
        for (int kk = 0; kk < 32; kk += 4) {
            const int klo = kk + lhalf * 2;
            v2f a0, a1, b0, b1;
            a0.x = As[(wr + lidx) * 33 + klo];       a0.y = As[(wr + lidx) * 33 + klo + 1];
            a1.x = As[(wr + 16 + lidx) * 33 + klo];  a1.y = As[(wr + 16 + lidx) * 33 + klo + 1];
            b0.x = Bs[(wc + lidx) * 33 + klo];       b0.y = Bs[(wc + lidx) * 33 + klo + 1];
            b1.x = Bs[(wc + 16 + lidx) * 33 + klo];  b1.y = Bs[(wc + 16 + lidx) * 33 + klo + 1];
            acc00 = __builtin_amdgcn_wmma_f32_16x16x4_f32(false, a0, false, b0, (short)0, acc00, false, false);
            acc01 = __builtin_amdgcn_wmma_f32_16x16x4_f32(false, a0, false, b1, (short)0, acc01, false, false);
            acc10 = __builtin_amdgcn_wmma_f32_16x16x4_f32(false, a1, false, b0, (short)0, acc10, false, false);
            acc11 = __builtin_amdgcn_wmma_f32_16x16x4_f32(false, a1, false, b1, (short)0, acc11, false, false);
        }
    }

    // epilogue (C/D layout: vgpr v -> M = v + 8*lhalf, N = lidx)
    #pragma unroll
    for (int i = 0; i < 2; ++i) {
        #pragma unroll
        for (int j = 0; j < 2; ++j) {
            v8f acc = (i == 0) ? (j == 0 ? acc00 : acc01) : (j == 0 ? acc10 : acc11);
            #pragma unroll
            for (int v = 0; v < 8; ++v) {
                int row = row0 + wr + i * 16 + v + lhalf * 8;
                int col = col0 + wc + j * 16 + lidx;
                if (row < Mrows && col < Nout) {
                    float val = acc[v];
                    if (bias)   val += bias[col];
                    if (addend) val += addend[(size_t)row * Nout + col];
                    if (act == 1)      val = tanhf(val);
                    else if (act == 2) val = 1.f / (1.f + expf(-val));
                    if (mul)    val *= mul[(size_t)row * Nout + col];
                    Y[(size_t)row * Nout + col] = val;
                }
            }
        }
    }
}

// ---------------------------------------------------------------------------
// Chunked EMA:  h_t = alpha*h_{t-1} + b_t  (alpha = exp(-softplus(tau)))
// pass 1: per-chunk local scan; pass 2: scan chunk carries; pass 3: fixup.
// ---------------------------------------------------------------------------
__global__ __launch_bounds__(256)
void ema_local(const float* __restrict__ in, float* __restrict__ out,
               float* __restrict__ carry, const float* __restrict__ tau)
{
    int c = blockIdx.x * 256 + threadIdx.x;
    int ch = blockIdx.y, b = blockIdx.z;
    float sp = log1pf(expf(tau[0]));
    float alpha = expf(-sp);
    float h = 0.f;
    size_t base = ((size_t)(b * Tt + ch * CHk)) * Cc + c;
    for (int t = 0; t < CHk; ++t) {
        h = fmaf(alpha, h, in[base + (size_t)t * Cc]);
        out[base + (size_t)t * Cc] = h;
    }
    carry[((size_t)b * NCH + ch) * Cc + c] = h;
}

__global__ __launch_bounds__(256)
void ema_scan(float* __restrict__ carry, const float* __restrict__ tau)
{
    int c = blockIdx.x * 256 + threadIdx.x;
    int b = blockIdx.y;
    float sp  = log1pf(expf(tau[0]));
    float aCH = expf(-sp * (float)CHk);
    float H = 0.f;
    for (int ch = 0; ch < NCH; ++ch) {
        size_t idx = ((size_t)b * NCH + ch) * Cc + c;
        H = carry[idx] + aCH * H;
        carry[idx] = H;
    }
}

__global__ __launch_bounds__(256)
void ema_fixup(float* __restrict__ out, const float* __restrict__ carry,
               const float* __restrict__ tau)
{
    int c = blockIdx.x * 256 + threadIdx.x;
    int t = blockIdx.y, b = blockIdx.z;
    if (t < CHk) return;
    int ch = t / CHk, tloc = t % CHk;
    float sp = log1pf(expf(tau[0]));
    float scale = expf(-sp * (float)(tloc + 1));
    float Hp = carry[((size_t)b * NCH + (ch - 1)) * Cc + c];
    size_t idx = ((size_t)(b * Tt + t)) * Cc + c;
    out[idx] += scale * Hp;
}

// ---------------------------------------------------------------------------
// Content gate: cg[b,t] = sigmoid( max_{s<=t} q_t . k_s / sqrt(DK) )
// Block = (64 t-rows) x (4 s-partitions); LDS tiles with pitch 100 (conflict-free).
// ---------------------------------------------------------------------------
__global__ __launch_bounds__(256)
void cg_kernel(const float* __restrict__ Q, const float* __restrict__ K,
               float* __restrict__ cg)
{
    __shared__ float qs[64 * 100];
    __shared__ float ks[64 * 100];
    __shared__ float red[256];

    int tid = threadIdx.x;
    int b = blockIdx.y;
    int t0 = blockIdx.x * 64;

    for (int idx = tid; idx < 64 * DKk; idx += 256) {
        int r = idx / DKk, k = idx % DKk;
        qs[r * 100 + k] = Q[((size_t)(b * Tt) + t0 + r) * DKk + k];
    }

    int ty = tid >> 2, tx = tid & 3;
    float m = -INFINITY;

    for (int s0 = 0; s0 <= t0; s0 += 64) {
        __syncthreads();
        for (int idx = tid; idx < 64 * DKk; idx += 256) {
            int r = idx / DKk, k = idx % DKk;
            ks[r * 100 + k] = K[((size_t)(b * Tt) + s0 + r) * DKk + k];
        }
        __syncthreads();
        bool diag = (s0 == t0);
        for (int j = 0; j < 16; ++j) {
            int sl = tx * 16 + j;
            if (diag && sl > ty) continue;   // causal mask on diagonal tile
            float d = 0.f;
            #pragma unroll 4
            for (int k = 0; k < DKk; ++k)
                d = fmaf(qs[ty * 100 + k], ks[sl * 100 + k], d);
            m = fmaxf(m, d);
        }
    }
    red[tid] = m;
    __syncthreads();
    if (tx == 0) {
        float mm = fmaxf(fmaxf(red[tid], red[tid + 1]), fmaxf(red[tid + 2], red[tid + 3]));
        cg[(size_t)b * Tt + t0 + ty] = 1.f / (1.f + expf(-mm * 0.1020620726159658f));
    }
}

// ---------------------------------------------------------------------------
// Anchors: delta[b,c] = sum_n sigmoid(scales[n]) * tanh(x[b,pos_n,:].Wanc[c,:] + banc[c])
// ---------------------------------------------------------------------------
__global__ __launch_bounds__(256)
void anchor_kernel(const float* __restrict__ x, const float* __restrict__ Wanc,
                   const float* __restrict__ banc, const float* __restrict__ scales,
                   float* __restrict__ delta)
{
    __shared__ float xr[Nanc * Cc];
    int b = blockIdx.x, tid = threadIdx.x;
    for (int idx = tid; idx < Nanc * Cc; idx += 256) {
        int n = idx / Cc, k = idx % Cc;
        int pos = n * (Tt / Nanc); if (pos > Tt - 1) pos = Tt - 1;
        xr[idx] = x[((size_t)(b * Tt + pos)) * Cc + k];
    }
    __syncthreads();
    for (int c = tid; c < Cc; c += 256) {
        float d = 0.f;
        const float* w = Wanc + (size_t)c * Cc;
        for (int n = 0; n < Nanc; ++n) {
            float a = banc[c];
            for (int k = 0; k < Cc; ++k) a = fmaf(xr[n * Cc + k], w[k], a);
            float wn = 1.f / (1.f + expf(-scales[n]));
            d += wn * tanhf(a);
        }
        delta[(size_t)b * Cc + c] = d;
    }
}

// ---------------------------------------------------------------------------
// ctx adjust: U = U * cg[b,t] + delta[b,c]  (both fast & slow)
// ---------------------------------------------------------------------------
__global__ __launch_bounds__(256)
void adjust_kernel(float* __restrict__ UF, float* __restrict__ US,
                   const float* __restrict__ cg, const float* __restrict__ delta)
{
    int c = blockIdx.x * 256 + threadIdx.x;
    int t = blockIdx.y, b = blockIdx.z;
    float g = cg[(size_t)b * Tt + t];
    float d = delta[(size_t)b * Cc + c];
    size_t idx = ((size_t)(b * Tt + t)) * Cc + c;
    UF[idx] = UF[idx] * g + d;
    US[idx] = US[idx] * g + d;
}

// ---------------------------------------------------------------------------
// Final: gate = lerp(gf, gs, bl); y = soma*gate; LayerNorm over C.
// ---------------------------------------------------------------------------
__global__ __launch_bounds__(256)
void final_ln(const float* __restrict__ GF, const float* __restrict__ GS,
              const float* __restrict__ SOMA, const float* __restrict__ blend,
              const float* __restrict__ lng, const float* __restrict__ lnb,
              float* __restrict__ out)
{
    __shared__ float s1[256], s2[256];
    int r = blockIdx.x, tid = threadIdx.x;
    float bl = 1.f / (1.f + expf(-blend[0]));
    float yv[3];
    float su = 0.f, sq = 0.f;
    #pragma unroll
    for (int i = 0; i < 3; ++i) {
        int c = tid + i * 256;
        size_t idx = (size_t)r * Cc + c;
        float gf = GF[idx], gs = GS[idx], so = SOMA[idx];
        float y = so * (gf + bl * (gs - gf));
        yv[i] = y; su += y; sq += y * y;
    }
    s1[tid] = su; s2[tid] = sq;
    __syncthreads();
    for (int off = 128; off > 0; off >>= 1) {
        if (tid < off) { s1[tid] += s1[tid + off]; s2[tid] += s2[tid + off]; }
        __syncthreads();
    }
    float mu  = s1[0] / (float)Cc;
    float var = s2[0] / (float)Cc - mu * mu;
    float inv = rsqrtf(var + 1e-5f);
    #pragma unroll
    for (int i = 0; i < 3; ++i) {
        int c = tid + i * 256;
        out[(size_t)r * Cc + c] = (yv[i] - mu) * inv * lng[c] + lnb[c];
    }
}

// ---------------------------------------------------------------------------
extern "C" void kernel_launch(void* const* d_in, const int* in_sizes, int n_in,
                              void* d_out, int out_size, void* d_ws, size_t ws_size,
                              hipStream_t stream)
{
    (void)in_sizes; (void)n_in; (void)out_size; (void)ws_size;
    const float* x        = (const float*)d_in[0];
    const float* tau_gate = (const float*)d_in[1];
    const float* tau_fast = (const float*)d_in[2];
    const float* tau_slow = (const float*)d_in[3];
    const float* Wp   = (const float*)d_in[4];  const float* bp    = (const float*)d_in[5];
    const float* Wxf  = (const float*)d_in[6];  const float* bxf   = (const float*)d_in[7];
    const float* Wxs  = (const float*)d_in[8];  const float* bxs   = (const float*)d_in[9];
    const float* Wcf  = (const float*)d_in[10]; const float* Wcs   = (const float*)d_in[11];
    const float* Wq   = (const float*)d_in[12]; const float* Wk    = (const float*)d_in[13];
    const float* Wfast= (const float*)d_in[14]; const float* bfast = (const float*)d_in[15];
    const float* Wslow= (const float*)d_in[16]; const float* bslow = (const float*)d_in[17];
    const float* Wsoma= (const float*)d_in[18]; const float* bsoma = (const float*)d_in[19];
    const float* Wanc = (const float*)d_in[20]; const float* banc  = (const float*)d_in[21];
    const float* ascl = (const float*)d_in[22];
    const float* blend= (const float*)d_in[23];
    const float* lng  = (const float*)d_in[24]; const float* lnb   = (const float*)d_in[25];
    float* out = (float*)d_out;

    float* ws = (float*)d_ws;
    const size_t SZ = (size_t)Mm * Cc;
    float* P   = ws + 0 * SZ;   // x_proj ; later reused as GF
    float* XF  = ws + 1 * SZ;   // x@Wxf^T + bxf ; later reused as GS
    float* XS  = ws + 2 * SZ;
    float* G   = ws + 3 * SZ;   // ctx_gate ; later reused as SOMA
    float* UF  = ws + 4 * SZ;   // u_f -> ctx_fast (in place)
    float* US  = ws + 5 * SZ;
    float* Qb  = ws + 6 * SZ;
    float* Kb    = Qb + (size_t)Mm * DKk;
    float* cgb   = Kb + (size_t)Mm * DKk;
    float* dl    = cgb + Mm;
    float* carry = dl + (size_t)Bx * Cc;
    float* SOMA = G;
    float* GF   = P;
    float* GS   = XF;

    dim3 blk(256);
    dim3 gG(Cc / 64, Mm / 128);            // 768-wide GEMMs
    dim3 gQK((DKk + 63) / 64, Mm / 128);   // 96-wide GEMMs
    dim3 gEW(Cc / 256, Tt, Bx);            // elementwise over (b,t,c)
    dim3 gEL(Cc / 256, NCH, Bx);
    dim3 gES(Cc / 256, Bx);

    // GEMMs from x
    gemm_wmma_f32<<<gG,  blk, 0, stream>>>(x, Wp,  bp,  nullptr, nullptr, P,  Mm, Cc,  Cc, 1); // x_proj = tanh
    gemm_wmma_f32<<<gG,  blk, 0, stream>>>(x, Wxf, bxf, nullptr, nullptr, XF, Mm, Cc,  Cc, 0);
    gemm_wmma_f32<<<gG,  blk, 0, stream>>>(x, Wxs, bxs, nullptr, nullptr, XS, Mm, Cc,  Cc, 0);
    gemm_wmma_f32<<<gQK, blk, 0, stream>>>(x, Wq,  nullptr, nullptr, nullptr, Qb, Mm, DKk, Cc, 0);
    gemm_wmma_f32<<<gQK, blk, 0, stream>>>(x, Wk,  nullptr, nullptr, nullptr, Kb, Mm, DKk, Cc, 0);

    // ctx_gate = EMA(x, tau_gate)
    ema_local<<<gEL, blk, 0, stream>>>(x, G, carry, tau_gate);
    ema_scan <<<gES, blk, 0, stream>>>(carry, tau_gate);
    ema_fixup<<<gEW, blk, 0, stream>>>(G, carry, tau_gate);

    // u_f = sigmoid(G@Wcf^T + XF) * x_proj ;  u_s likewise
    gemm_wmma_f32<<<gG, blk, 0, stream>>>(G, Wcf, nullptr, XF, P, UF, Mm, Cc, Cc, 2);
    gemm_wmma_f32<<<gG, blk, 0, stream>>>(G, Wcs, nullptr, XS, P, US, Mm, Cc, Cc, 2);

    // soma = tanh(x@Wsoma^T + bsoma)  (reuses G slot)
    gemm_wmma_f32<<<gG, blk, 0, stream>>>(x, Wsoma, bsoma, nullptr, nullptr, SOMA, Mm, Cc, Cc, 1);

    // ctx_fast / ctx_slow EMAs (in place)
    ema_local<<<gEL, blk, 0, stream>>>(UF, UF, carry, tau_fast);
    ema_scan <<<gES, blk, 0, stream>>>(carry, tau_fast);
    ema_fixup<<<gEW, blk, 0, stream>>>(UF, carry, tau_fast);
    ema_local<<<gEL, blk, 0, stream>>>(US, US, carry, tau_slow);
    ema_scan <<<gES, blk, 0, stream>>>(carry, tau_slow);
    ema_fixup<<<gEW, blk, 0, stream>>>(US, carry, tau_slow);

    // content gate + anchors + adjust
    cg_kernel<<<dim3(Tt / 64, Bx), blk, 0, stream>>>(Qb, Kb, cgb);
    anchor_kernel<<<dim3(Bx), blk, 0, stream>>>(x, Wanc, banc, ascl, dl);
    adjust_kernel<<<gEW, blk, 0, stream>>>(UF, US, cgb, dl);

    // gate GEMMs (reuse P / XF slots)
    gemm_wmma_f32<<<gG, blk, 0, stream>>>(UF, Wfast, bfast, nullptr, nullptr, GF, Mm, Cc, Cc, 2);
    gemm_wmma_f32<<<gG, blk, 0, stream>>>(US, Wslow, bslow, nullptr, nullptr, GS, Mm, Cc, Cc, 2);

    // blend + soma mul + layernorm
    final_ln<<<dim3(Mm), blk, 0, stream>>>(GF, GS, SOMA, blend, lng, lnb, out);
}